// GraphNeuralSurrogate_11493332484599
// MI455X (gfx1250) — compile-verified
//
#include <hip/hip_runtime.h>
#include <hip/hip_bf16.h>

typedef float v2f __attribute__((ext_vector_type(2)));
typedef float v8f __attribute__((ext_vector_type(8)));

#define HIDDEN 64
#define HSTR   68          // padded LDS row stride in floats (conflict-free A loads)
#define WAVES  8           // waves per 256-thread block (wave32)
#define TILE_M 16

// ---- packed-weight (d_ws) layout, in floats ----
// Fragment order: for each K-step s, lane L (h=L>>4, lm=L&15) holds its
// B-fragment values contiguously: j = 2*t + v  ->  W[k = 4s+2h+v][n = 16t+lm]
#define SWZ_WIN   0                    // 32 lanes * 8  = 256   (W_in, 4x64, s=0)
#define SWZ_WS    256                  // 3*16 steps * 32 lanes * 8 = 12288 (Ws)
#define SWZ_W1    (256 + 12288)        // 16 steps * 32 lanes * 4 = 2048   (W1, 64x32)
#define SWZ_TOTAL 14592                // floats (58368 bytes)

__device__ __forceinline__ v8f wmma4(v2f a, v2f b, v8f c) {
  // V_WMMA_F32_16X16X4_F32: D(16x16 f32) = A(16x4 f32) * B(4x16 f32) + C
  return __builtin_amdgcn_wmma_f32_16x16x4_f32(false, a, false, b, (short)0, c, false, false);
}

// One-time (per launch) weight swizzle into fragment-packed order.
__global__ __launch_bounds__(256) void swizzle_weights_kernel(
    const float* __restrict__ W_in,   // (4,64)
    const float* __restrict__ Ws,     // (3,64,64)
    const float* __restrict__ W1,     // (64,32)
    float* __restrict__ swz)
{
  const int i = blockIdx.x * blockDim.x + threadIdx.x;
  if (i >= SWZ_TOTAL) return;
  if (i < SWZ_WS) {                          // W_in region
    const int j = i & 7, lane = (i >> 3) & 31;
    const int h = lane >> 4, lm = lane & 15;
    const int k = 2 * h + (j & 1), n = 16 * (j >> 1) + lm;
    swz[i] = W_in[k * HIDDEN + n];
  } else if (i < SWZ_W1) {                   // Ws region
    const int r = i - SWZ_WS;
    const int j = r & 7, lane = (r >> 3) & 31, ls = r >> 8;   // ls = l*16 + s
    const int s = ls & 15, l = ls >> 4;
    const int h = lane >> 4, lm = lane & 15;
    const int k = 4 * s + 2 * h + (j & 1), n = 16 * (j >> 1) + lm;
    swz[i] = Ws[l * HIDDEN * HIDDEN + k * HIDDEN + n];
  } else {                                   // W1 region
    const int r = i - SWZ_W1;
    const int j = r & 3, lane = (r >> 2) & 31, s = r >> 7;
    const int h = lane >> 4, lm = lane & 15;
    const int k = 4 * s + 2 * h + (j & 1), n = 16 * (j >> 1) + lm;
    swz[i] = W1[k * 32 + n];
  }
}

template <bool PACKED>
__global__ __launch_bounds__(256) void gnn_surrogate_kernel(
    const float* __restrict__ x,      // (B,4)
    const float* __restrict__ W_in,   // (4,64)
    const float* __restrict__ b_in,   // (64)
    const float* __restrict__ Ws,     // (3,64,64)
    const float* __restrict__ bs,     // (3,64)
    const float* __restrict__ W1,     // (64,32)
    const float* __restrict__ b1,     // (32)
    const float* __restrict__ W2,     // (32,1)
    const float* __restrict__ b2,     // (1)
    const float* __restrict__ swz,    // packed weights (or null)
    float* __restrict__ out,          // (B,1)
    int numTiles)
{
  __shared__ float lds[WAVES * TILE_M * HSTR];

  const int lane = threadIdx.x & 31;
  const int wave = threadIdx.x >> 5;
  const int tile = blockIdx.x * WAVES + wave;
  if (tile >= numTiles) return;               // wave-uniform: EXEC stays all-ones

  const int lm   = lane & 15;
  const int half = lane >> 4;
  float* hb = &lds[wave * TILE_M * HSTR];     // wave-private 16x64 (stride 68) tile

  // ---------------- input projection: h = x @ W_in + b_in  (K = 4)
  v2f A;
  {
    const float2 xv = *(const float2*)(x + (size_t)(tile * TILE_M + lm) * 4 + 2 * half);
    A.x = xv.x; A.y = xv.y;
  }
  v2f Bt[4];
  if constexpr (PACKED) {
    const float4 q0 = *(const float4*)(swz + SWZ_WIN + lane * 8);
    const float4 q1 = *(const float4*)(swz + SWZ_WIN + lane * 8 + 4);
    Bt[0] = (v2f){q0.x, q0.y}; Bt[1] = (v2f){q0.z, q0.w};
    Bt[2] = (v2f){q1.x, q1.y}; Bt[3] = (v2f){q1.z, q1.w};
  } else {
#pragma unroll
    for (int t = 0; t < 4; ++t) {
      Bt[t].x = W_in[(2 * half + 0) * HIDDEN + 16 * t + lm];
      Bt[t].y = W_in[(2 * half + 1) * HIDDEN + 16 * t + lm];
    }
  }
  v8f acc[4];
#pragma unroll
  for (int t = 0; t < 4; ++t) {
    v8f c = {};
    c = wmma4(A, Bt[t], c);
    const float bias = b_in[16 * t + lm];
#pragma unroll
    for (int v = 0; v < 8; ++v) c[v] += bias;
    acc[t] = c;                               // no relu after input projection
  }

  // ---------------- 3 GCN layers; aggregation over identical nodes == identity
#pragma unroll
  for (int l = 0; l < 3; ++l) {
    const float* __restrict__ W = Ws + (size_t)l * HIDDEN * HIDDEN;

    // spill h (D layout) to LDS row-major so we can re-read A fragments
#pragma unroll
    for (int t = 0; t < 4; ++t)
#pragma unroll
      for (int v = 0; v < 8; ++v)
        hb[(v + 8 * half) * HSTR + 16 * t + lm] = acc[t][v];
    __builtin_amdgcn_wave_barrier();          // DS ops are in-order per wave

    v8f nacc[4];
#pragma unroll
    for (int t = 0; t < 4; ++t) nacc[t] = (v8f){};

#pragma unroll
    for (int s = 0; s < 16; ++s) {            // K-steps of 4
      v2f Af;
      {
        const float2 av = *(const float2*)&hb[lm * HSTR + 4 * s + 2 * half];
        Af.x = av.x; Af.y = av.y;
      }
      v2f Bw[4];
      if constexpr (PACKED) {
        const float* p = swz + SWZ_WS + ((l * 16 + s) * 32 + lane) * 8;
        const float4 q0 = *(const float4*)(p);
        const float4 q1 = *(const float4*)(p + 4);
        Bw[0] = (v2f){q0.x, q0.y}; Bw[1] = (v2f){q0.z, q0.w};
        Bw[2] = (v2f){q1.x, q1.y}; Bw[3] = (v2f){q1.z, q1.w};
      } else {
#pragma unroll
        for (int t = 0; t < 4; ++t) {
          Bw[t].x = W[(4 * s + 2 * half + 0) * HIDDEN + 16 * t + lm];
          Bw[t].y = W[(4 * s + 2 * half + 1) * HIDDEN + 16 * t + lm];
        }
      }
#pragma unroll
      for (int t = 0; t < 4; ++t) nacc[t] = wmma4(Af, Bw[t], nacc[t]);
    }
#pragma unroll
    for (int t = 0; t < 4; ++t) {
      const float bias = bs[l * HIDDEN + 16 * t + lm];   // zeros, kept for fidelity
      v8f c = nacc[t];
#pragma unroll
      for (int v = 0; v < 8; ++v) c[v] = fmaxf(c[v] + bias, 0.0f);
      acc[t] = c;
    }
    __builtin_amdgcn_wave_barrier();
  }

  // ---------------- head: z = relu(h @ W1 + b1)   (64 -> 32)
#pragma unroll
  for (int t = 0; t < 4; ++t)
#pragma unroll
    for (int v = 0; v < 8; ++v)
      hb[(v + 8 * half) * HSTR + 16 * t + lm] = acc[t][v];
  __builtin_amdgcn_wave_barrier();

  v8f z0 = {}, z1 = {};
#pragma unroll
  for (int s = 0; s < 16; ++s) {
    v2f Af;
    {
      const float2 av = *(const float2*)&hb[lm * HSTR + 4 * s + 2 * half];
      Af.x = av.x; Af.y = av.y;
    }
    v2f B0, B1;
    if constexpr (PACKED) {
      const float4 q = *(const float4*)(swz + SWZ_W1 + (s * 32 + lane) * 4);
      B0 = (v2f){q.x, q.y}; B1 = (v2f){q.z, q.w};
    } else {
      B0.x = W1[(4 * s + 2 * half + 0) * 32 + lm];
      B0.y = W1[(4 * s + 2 * half + 1) * 32 + lm];
      B1.x = W1[(4 * s + 2 * half + 0) * 32 + 16 + lm];
      B1.y = W1[(4 * s + 2 * half + 1) * 32 + 16 + lm];
    }
    z0 = wmma4(Af, B0, z0);
    z1 = wmma4(Af, B1, z1);
  }

  // bias + relu, park z (16x32) back in LDS
  {
    const float bb0 = b1[lm];
    const float bb1 = b1[16 + lm];
#pragma unroll
    for (int v = 0; v < 8; ++v) {
      hb[(v + 8 * half) * HSTR + lm]      = fmaxf(z0[v] + bb0, 0.0f);
      hb[(v + 8 * half) * HSTR + 16 + lm] = fmaxf(z1[v] + bb1, 0.0f);
    }
  }
  __builtin_amdgcn_wave_barrier();

  // ---------------- out = z @ W2 + b2   (32 -> 1); split j-range across halves
  float partial = 0.0f;
#pragma unroll
  for (int j = 0; j < 16; ++j)
    partial += hb[lm * HSTR + 16 * half + j] * W2[16 * half + j];
  partial += __shfl_xor(partial, 16, 32);     // combine the two halves

  if (half == 0)
    out[tile * TILE_M + lm] = partial + b2[0];
}

extern "C" void kernel_launch(void* const* d_in, const int* in_sizes, int n_in,
                              void* d_out, int out_size, void* d_ws, size_t ws_size,
                              hipStream_t stream) {
  const float* x    = (const float*)d_in[0];
  const float* W_in = (const float*)d_in[1];
  const float* b_in = (const float*)d_in[2];
  const float* Ws   = (const float*)d_in[3];
  const float* bs   = (const float*)d_in[4];
  const float* W1   = (const float*)d_in[5];
  const float* b1   = (const float*)d_in[6];
  const float* W2   = (const float*)d_in[7];
  const float* b2   = (const float*)d_in[8];
  float* out = (float*)d_out;

  const int batch    = in_sizes[0] / 4;                 // x is (B,4)
  const int numTiles = (batch + TILE_M - 1) / TILE_M;   // 16384 for B=262144
  const int blocks   = (numTiles + WAVES - 1) / WAVES;  // 8 waves/block, 1 tile/wave

  const bool packed = (d_ws != nullptr) && (ws_size >= SWZ_TOTAL * sizeof(float));
  if (packed) {
    float* swz = (float*)d_ws;
    swizzle_weights_kernel<<<(SWZ_TOTAL + 255) / 256, 256, 0, stream>>>(W_in, Ws, W1, swz);
    gnn_surrogate_kernel<true><<<blocks, 256, 0, stream>>>(
        x, W_in, b_in, Ws, bs, W1, b1, W2, b2, swz, out, numTiles);
  } else {
    gnn_surrogate_kernel<false><<<blocks, 256, 0, stream>>>(
        x, W_in, b_in, Ws, bs, W1, b1, W2, b2, nullptr, out, numTiles);
  }
}